// EF_ChargeSpinConditioned_3685081940449
// MI455X (gfx1250) — compile-verified
//
#include <hip/hip_runtime.h>
#include <hip/hip_bf16.h>

typedef float v2f __attribute__((ext_vector_type(2)));
typedef float v8f __attribute__((ext_vector_type(8)));

#define F 32
#define NB 16

__device__ __forceinline__ float fast_silu(float x) {
    // x * sigmoid(x); v_exp_f32 + v_rcp_f32 (both TRANS, co-execute with WMMA)
    return x * __builtin_amdgcn_rcpf(1.0f + __expf(-x));
}

// ---------------------------------------------------------------- zero scratch
__global__ void zero_kernel(float* __restrict__ p, int n) {
    int g = blockIdx.x * blockDim.x + threadIdx.x;
    if (g < n) p[g] = 0.0f;
}

// ---------------------------------------------------------------- mol embedding
// mol[b,f] = concat(charge_embed[ci], spin_embed[si]) @ mol_W + mol_b
__global__ void mol_kernel(const int* __restrict__ q, const int* __restrict__ sp,
                           const float* __restrict__ ce, const float* __restrict__ se,
                           const float* __restrict__ mW, const float* __restrict__ mb,
                           float* __restrict__ mol, int B) {
    int g = blockIdx.x * blockDim.x + threadIdx.x;
    if (g >= B * F) return;
    int b = g >> 5, f = g & 31;
    int ci = min(max(q[b] + 5, 0), 10);
    int si = min(max(sp[b] - 1, 0), 6);
    float acc = mb[f];
    #pragma unroll
    for (int k = 0; k < 16; ++k) acc += ce[ci * 16 + k] * mW[k * F + f];
    #pragma unroll
    for (int k = 0; k < 16; ++k) acc += se[si * 16 + k] * mW[(16 + k) * F + f];
    mol[g] = acc;
}

// ---------------------------------------------------------------- x0 = atom_embed[Z] + mol[seg]
__global__ void x0_kernel(const int* __restrict__ Z, const int* __restrict__ seg,
                          const float* __restrict__ ae, const float* __restrict__ mol,
                          float* __restrict__ x0, int A) {
    int g = blockIdx.x * blockDim.x + threadIdx.x;
    if (g >= A * F) return;
    int a = g >> 5, f = g & 31;
    x0[g] = ae[Z[a] * F + f] + mol[seg[a] * F + f];
}

// ---------------------------------------------------------------- edge message pass
// thread (e, f): scale = cutv * (Cheb(r)@Wb[:,f] + add_cut); accum[dst,f] += scale * xin[src,f]
__global__ __launch_bounds__(256) void edge_kernel(
    const float* __restrict__ pos, const int* __restrict__ dst,
    const int* __restrict__ src, const float* __restrict__ Wb,
    const float* __restrict__ xin, float* __restrict__ accum,
    int P, int add_cut) {
    long long g = (long long)blockIdx.x * blockDim.x + threadIdx.x;
    int e = (int)(g >> 5);
    int f = (int)(g & 31);
    if (e >= P) return;
    int s = src[e], d = dst[e];
    float dx = pos[s * 3 + 0] - pos[d * 3 + 0];
    float dy = pos[s * 3 + 1] - pos[d * 3 + 1];
    float dz = pos[s * 3 + 2] - pos[d * 3 + 2];
    float r = sqrtf(dx * dx + dy * dy + dz * dz);
    // smooth cutoff
    float u = r * (1.0f / 6.0f);
    u = u * u;
    float cutv = 0.0f;
    if (u < 1.0f)
        cutv = __expf(1.0f - __builtin_amdgcn_rcpf(fmaxf(1.0f - u, 1e-6f)));
    // exponential-Chebyshev radial dotted with Wb[:, f]
    float zc = fminf(fmaxf(2.0f * __expf(-r) - 1.0f, -1.0f), 1.0f);
    float Tm2 = 1.0f, Tm1 = zc;
    float c = Wb[f] + Wb[F + f] * zc;
    #pragma unroll
    for (int b2 = 2; b2 < NB; ++b2) {
        float T = 2.0f * zc * Tm1 - Tm2;
        c += Wb[b2 * F + f] * T;
        Tm2 = Tm1; Tm1 = T;
    }
    float scale = cutv * (c + (add_cut ? 1.0f : 0.0f));
    float val = scale * xin[s * F + f];
    unsafeAtomicAdd(&accum[d * F + f], val);  // global_atomic_add_f32
}

// ---------------------------------------------------------------- residual-MLP refine (WMMA)
// x = xa + xb (per row of 32 feats); 3x { h = silu(x@W1+b1); y = h@W2+b2; x = silu(x+y) }
// One wave = one 16-row tile; 256 threads = 8 waves = 128 rows per block.
// Weights staged into LDS in fragment-interleaved form:
//   Wf[(k>>1)*64 + n*2 + (k&1)] = W[k][n]
// so every B-fragment is a single contiguous 8B-aligned ds_load_b64.
__global__ __launch_bounds__(256) void refine_kernel(
    const float* __restrict__ xa, const float* __restrict__ xb,
    const float* __restrict__ W1s, const float* __restrict__ b1s,
    const float* __restrict__ W2s, const float* __restrict__ b2s,
    float* __restrict__ out, int A) {
    __shared__ __align__(16) float sW1[3 * F * F];
    __shared__ __align__(16) float sW2[3 * F * F];
    __shared__ float sb1[3 * F];
    __shared__ float sb2[3 * F];
    __shared__ __align__(16) float xt[128 * F];
    __shared__ __align__(16) float ht[128 * F];

    const int t = threadIdx.x;
    const int lane = t & 31;
    const int wave = t >> 5;
    const int base = blockIdx.x * 128 * F;
    const int lim = A * F;

    // fragment-interleaved weight staging
    for (int i = t; i < 3 * F * F; i += 256) {
        int rem = i & 1023;
        int k = rem >> 5;
        int n = rem & 31;
        int fidx = (i & ~1023) + ((k >> 1) << 6) + (n << 1) + (k & 1);
        sW1[fidx] = W1s[i];
        sW2[fidx] = W2s[i];
    }
    for (int i = t; i < 3 * F; i += 256) { sb1[i] = b1s[i]; sb2[i] = b2s[i]; }
    for (int i = t; i < 128 * F; i += 256)
        xt[i] = (base + i < lim) ? (xa[base + i] + xb[base + i]) : 0.0f;
    __syncthreads();

    // WMMA f32 16x16x4 fragment coordinates (wave32)
    const int m     = lane & 15;            // A-frag row
    const int khalf = (lane >> 4) << 1;     // A/B K sub-pair: 0 or 2
    const int row0  = wave * 16;            // this wave's tile row base
    const int cm    = (lane >> 4) << 3;     // C-frag M base: 0 or 8
    const int cn    = lane & 15;            // C-frag N (within 16-col tile)

    for (int j = 0; j < 3; ++j) {
        const v2f* W1p = (const v2f*)(sW1 + j * F * F);  // [kpair*32 + n]
        const v2f* W2p = (const v2f*)(sW2 + j * F * F);
        const float* B1 = sb1 + j * F;
        const float* B2 = sb2 + j * F;

        // ---- h = silu(x @ W1 + b1)
        v8f acc0, acc1;
        {
            float bn0 = B1[cn], bn1 = B1[cn + 16];
            #pragma unroll
            for (int q = 0; q < 8; ++q) { acc0[q] = bn0; acc1[q] = bn1; }
        }
        #pragma unroll
        for (int kc = 0; kc < 8; ++kc) {
            int kb = kc * 4 + khalf;
            int kp = (kb >> 1) << 5;
            v2f a  = *(const v2f*)(xt + (row0 + m) * F + kb);
            v2f b0 = W1p[kp + cn];
            v2f b1 = W1p[kp + cn + 16];
            acc0 = __builtin_amdgcn_wmma_f32_16x16x4_f32(false, a, false, b0, (short)0, acc0, false, false);
            acc1 = __builtin_amdgcn_wmma_f32_16x16x4_f32(false, a, false, b1, (short)0, acc1, false, false);
        }
        #pragma unroll
        for (int q = 0; q < 8; ++q) {
            ht[(row0 + cm + q) * F + cn]      = fast_silu(acc0[q]);
            ht[(row0 + cm + q) * F + cn + 16] = fast_silu(acc1[q]);
        }
        __syncthreads();

        // ---- y = h @ W2 + b2 ; x = silu(x + y)
        {
            float bn0 = B2[cn], bn1 = B2[cn + 16];
            #pragma unroll
            for (int q = 0; q < 8; ++q) { acc0[q] = bn0; acc1[q] = bn1; }
        }
        #pragma unroll
        for (int kc = 0; kc < 8; ++kc) {
            int kb = kc * 4 + khalf;
            int kp = (kb >> 1) << 5;
            v2f a  = *(const v2f*)(ht + (row0 + m) * F + kb);
            v2f b0 = W2p[kp + cn];
            v2f b1 = W2p[kp + cn + 16];
            acc0 = __builtin_amdgcn_wmma_f32_16x16x4_f32(false, a, false, b0, (short)0, acc0, false, false);
            acc1 = __builtin_amdgcn_wmma_f32_16x16x4_f32(false, a, false, b1, (short)0, acc1, false, false);
        }
        __syncthreads();
        #pragma unroll
        for (int q = 0; q < 8; ++q) {
            float x0v = xt[(row0 + cm + q) * F + cn]      + acc0[q];
            float x1v = xt[(row0 + cm + q) * F + cn + 16] + acc1[q];
            xt[(row0 + cm + q) * F + cn]      = fast_silu(x0v);
            xt[(row0 + cm + q) * F + cn + 16] = fast_silu(x1v);
        }
        __syncthreads();
    }

    for (int i = t; i < 128 * F; i += 256)
        if (base + i < lim) out[base + i] = xt[i];
}

// ---------------------------------------------------------------- launch
extern "C" void kernel_launch(void* const* d_in, const int* in_sizes, int n_in,
                              void* d_out, int out_size, void* d_ws, size_t ws_size,
                              hipStream_t stream) {
    const float* pos  = (const float*)d_in[0];
    const int*   q    = (const int*)d_in[1];
    const int*   sp   = (const int*)d_in[2];
    const int*   Z    = (const int*)d_in[3];
    const int*   dst  = (const int*)d_in[4];
    const int*   src  = (const int*)d_in[5];
    const int*   seg  = (const int*)d_in[6];
    const float* ce   = (const float*)d_in[7];
    const float* se   = (const float*)d_in[8];
    const float* ae   = (const float*)d_in[9];
    const float* mW   = (const float*)d_in[10];
    const float* mb   = (const float*)d_in[11];
    const float* mpWb = (const float*)d_in[12];
    const float* rW1  = (const float*)d_in[13];
    const float* rb1  = (const float*)d_in[14];
    const float* rW2  = (const float*)d_in[15];
    const float* rb2  = (const float*)d_in[16];

    const int B = in_sizes[1];
    const int A = in_sizes[3];
    const int P = in_sizes[4];

    float* ws  = (float*)d_ws;
    float* mol = ws;                   // B*F
    float* x0  = mol + (size_t)B * F;  // A*F
    float* s0  = x0 + (size_t)A * F;   // A*F  (atomic accum, pass 1)
    float* xs0 = s0 + (size_t)A * F;   // A*F  (refined stage-0 result)
    float* s1  = xs0 + (size_t)A * F;  // A*F  (atomic accum, pass 2)

    // zero the two atomic accumulators (and xs0, overwritten anyway): 3*A*F from s0
    {
        int n = 3 * A * F;
        zero_kernel<<<(n + 255) / 256, 256, 0, stream>>>(s0, n);
    }
    mol_kernel<<<(B * F + 255) / 256, 256, 0, stream>>>(q, sp, ce, se, mW, mb, mol, B);
    x0_kernel<<<(A * F + 255) / 256, 256, 0, stream>>>(Z, seg, ae, mol, x0, A);

    const long long nedge = (long long)P * F;
    const int eblocks = (int)((nedge + 255) / 256);

    // pass 1: s0[dst] += (coeff0 + cutv) * x0[src]
    edge_kernel<<<eblocks, 256, 0, stream>>>(pos, dst, src, mpWb, x0, s0, P, 1);
    // refine stage 0: xs0 = refine(x0 + s0)
    refine_kernel<<<(A + 127) / 128, 256, 0, stream>>>(
        x0, s0, rW1, rb1, rW2, rb2, xs0, A);
    // pass 2: s1[dst] += coeff1 * xs0[src]
    edge_kernel<<<eblocks, 256, 0, stream>>>(pos, dst, src, mpWb + NB * F, xs0, s1, P, 0);
    // refine stage 1: out = refine(xs0 + s1)
    refine_kernel<<<(A + 127) / 128, 256, 0, stream>>>(
        xs0, s1, rW1 + 3 * F * F, rb1 + 3 * F, rW2 + 3 * F * F, rb2 + 3 * F,
        (float*)d_out, A);
}